// vaup_67345087201385
// MI455X (gfx1250) — compile-verified
//
#include <hip/hip_runtime.h>
#include <hip/hip_bf16.h>

typedef __attribute__((ext_vector_type(2))) float v2f;
typedef __attribute__((ext_vector_type(8))) float v8f;

#define BM 128
#define BN 128
#define KC 16
#define KP (KC / 2)          // k-pair rows held in LDS
#define LDS_S (2 * BM + 32)  // word stride per k-pair row; %64==32 -> half-waves on disjoint banks

// ---------------------------------------------------------------------------
// Kernel 1: coef[k] = sigmoid(x[k]·mean) * (1 - sigmoid) * (bs - k)
// One wave32 per row, float4 loads, shuffle reduction.
// ---------------------------------------------------------------------------
__global__ __launch_bounds__(256) void vaup_coef_kernel(
    const float* __restrict__ xt, const float* __restrict__ mean,
    float* __restrict__ coef, int bs, int d) {
  const int lane = threadIdx.x & 31;
  const int wave = threadIdx.x >> 5;
  const int row  = blockIdx.x * (blockDim.x >> 5) + wave;
  if (row >= bs) return;

  const float4* xr = reinterpret_cast<const float4*>(xt + (size_t)row * d);
  const float4* m4 = reinterpret_cast<const float4*>(mean);
  const int n4 = d >> 2;
  float s = 0.0f;
  for (int c = lane; c < n4; c += 32) {
    float4 xv = xr[c];
    float4 mv = m4[c];
    s += xv.x * mv.x + xv.y * mv.y + xv.z * mv.z + xv.w * mv.w;
  }
  #pragma unroll
  for (int off = 16; off > 0; off >>= 1) s += __shfl_xor(s, off, 32);
  if (lane == 0) {
    float p = 1.0f / (1.0f + __expf(-s));
    coef[row] = p * (1.0f - p) * (float)(bs - row);
  }
}

// ---------------------------------------------------------------------------
// Kernel 2: q = va + S, S = X^T * diag(coef) * X  via V_WMMA_F32_16X16X4_F32.
// S is symmetric: only upper-triangular 128x128 blocks are computed
// (nb*(nb+1)/2 blocks); off-diagonal blocks emit both S and S^T images.
// 256 threads (8 waves); waves 4(M) x 2(N); 2x4 WMMA tiles per wave.
// LDS is K-pair interleaved so one ds_load_b64 = one ready WMMA operand pair.
// Next chunk's global loads are register-prefetched during current math.
// ---------------------------------------------------------------------------
__global__ __launch_bounds__(256) void vaup_syrk_kernel(
    const float* __restrict__ xt, const float* __restrict__ va,
    const float* __restrict__ coef, float* __restrict__ q,
    int bs, int d, int nb) {
  __shared__ float XI2[KP * LDS_S];   // coef-scaled, i-columns
  __shared__ float XJ2[KP * LDS_S];   // raw, j-columns

  // Triangular decode: linear block id -> (bi, bj) with bi <= bj.
  int rem = blockIdx.x, bi = 0, rowlen = nb;
  while (rem >= rowlen) { rem -= rowlen; --rowlen; ++bi; }
  const int bj = bi + rem;

  const int t      = threadIdx.x;
  const int lane   = t & 31;
  const int waveId = t >> 5;
  const int wm     = waveId >> 1;   // 0..3 -> 32 output rows each
  const int wn     = waveId & 1;    // 0..1 -> 64 output cols each
  const int lhi    = lane >> 4;
  const int lcol   = lane & 15;

  const int iBase = bi * BM;
  const int jBase = bj * BN;

  // Staging slots: thread handles k-pairs kp0 and kp0+4, columns [c2, c2+1].
  const int kp0 = t >> 6;          // 0..3
  const int c2  = (t & 63) * 2;    // column start

  float2 pi[2][2], pj[2][2];       // [slot r][row parity]
  float  wreg[2][2];

  auto prefetch = [&](int kb) {
    #pragma unroll
    for (int r = 0; r < 2; ++r) {
      const int kp   = kp0 + r * 4;
      const int row0 = kb + 2 * kp;
      const size_t o0 = (size_t)row0 * d;
      pi[r][0] = *reinterpret_cast<const float2*>(xt + o0 + iBase + c2);
      pi[r][1] = *reinterpret_cast<const float2*>(xt + o0 + d + iBase + c2);
      pj[r][0] = *reinterpret_cast<const float2*>(xt + o0 + jBase + c2);
      pj[r][1] = *reinterpret_cast<const float2*>(xt + o0 + d + jBase + c2);
      wreg[r][0] = coef[row0];
      wreg[r][1] = coef[row0 + 1];
    }
  };

  auto stage = [&]() {
    #pragma unroll
    for (int r = 0; r < 2; ++r) {
      const int kp = kp0 + r * 4;
      const int w  = kp * LDS_S + c2 * 2;
      float4 vi, vj;
      vi.x = pi[r][0].x * wreg[r][0]; vi.y = pi[r][1].x * wreg[r][1];
      vi.z = pi[r][0].y * wreg[r][0]; vi.w = pi[r][1].y * wreg[r][1];
      vj.x = pj[r][0].x;              vj.y = pj[r][1].x;
      vj.z = pj[r][0].y;              vj.w = pj[r][1].y;
      *reinterpret_cast<float4*>(&XI2[w]) = vi;
      *reinterpret_cast<float4*>(&XJ2[w]) = vj;
    }
  };

  v8f acc[2][4];
  const v8f vzero = {0.f, 0.f, 0.f, 0.f, 0.f, 0.f, 0.f, 0.f};
  #pragma unroll
  for (int mt = 0; mt < 2; ++mt)
    #pragma unroll
    for (int nt = 0; nt < 4; ++nt) acc[mt][nt] = vzero;

  prefetch(0);

  for (int kb = 0; kb < bs; kb += KC) {
    __syncthreads();          // previous chunk's LDS fully consumed
    stage();
    __syncthreads();          // LDS tile visible to all waves
    if (kb + KC < bs) prefetch(kb + KC);   // overlap HBM/L2 latency with math

    #pragma unroll
    for (int k0 = 0; k0 < KC; k0 += 4) {
      const int base = ((k0 >> 1) + lhi) * LDS_S;   // half-wave k-pair select
      v2f a[2], b[4];
      #pragma unroll
      for (int mt = 0; mt < 2; ++mt)
        a[mt] = *reinterpret_cast<const v2f*>(&XI2[base + (wm * 32 + mt * 16 + lcol) * 2]);
      #pragma unroll
      for (int nt = 0; nt < 4; ++nt)
        b[nt] = *reinterpret_cast<const v2f*>(&XJ2[base + (wn * 64 + nt * 16 + lcol) * 2]);
      #pragma unroll
      for (int mt = 0; mt < 2; ++mt)
        #pragma unroll
        for (int nt = 0; nt < 4; ++nt)
          acc[mt][nt] = __builtin_amdgcn_wmma_f32_16x16x4_f32(
              false, a[mt], false, b[nt], (short)0, acc[mt][nt], false, false);
    }
  }

  // ---- Epilogue 1: direct image  q[i,j] = S + va[i,j].
  // C/D layout: m = r + 8*lhi, n = lcol.
  #pragma unroll
  for (int mt = 0; mt < 2; ++mt) {
    #pragma unroll
    for (int nt = 0; nt < 4; ++nt) {
      const int gn = jBase + wn * 64 + nt * 16 + lcol;
      #pragma unroll
      for (int r = 0; r < 8; ++r) {
        const int gm = iBase + wm * 32 + mt * 16 + r + 8 * lhi;
        const size_t off = (size_t)gm * d + gn;
        q[off] = acc[mt][nt][r] + va[off];
      }
    }
  }

  // ---- Epilogue 2: transposed image  q[j,i] = S^T + va[j,i]  (bi < bj only).
  // Per lane: output row = n (fixed), output cols = consecutive m -> float4 x2.
  if (bi != bj) {
    #pragma unroll
    for (int mt = 0; mt < 2; ++mt) {
      #pragma unroll
      for (int nt = 0; nt < 4; ++nt) {
        const int grow = jBase + wn * 64 + nt * 16 + lcol;          // q row
        const int gcol = iBase + wm * 32 + mt * 16 + 8 * lhi;       // q col start
        const size_t off = (size_t)grow * d + gcol;
        float4 v0 = *reinterpret_cast<const float4*>(va + off);
        float4 v1 = *reinterpret_cast<const float4*>(va + off + 4);
        float4 s0, s1;
        s0.x = acc[mt][nt][0] + v0.x; s0.y = acc[mt][nt][1] + v0.y;
        s0.z = acc[mt][nt][2] + v0.z; s0.w = acc[mt][nt][3] + v0.w;
        s1.x = acc[mt][nt][4] + v1.x; s1.y = acc[mt][nt][5] + v1.y;
        s1.z = acc[mt][nt][6] + v1.z; s1.w = acc[mt][nt][7] + v1.w;
        *reinterpret_cast<float4*>(q + off)     = s0;
        *reinterpret_cast<float4*>(q + off + 4) = s1;
      }
    }
  }
}

extern "C" void kernel_launch(void* const* d_in, const int* in_sizes, int n_in,
                              void* d_out, int out_size, void* d_ws, size_t ws_size,
                              hipStream_t stream) {
  const float* mean = (const float*)d_in[0];
  const float* va   = (const float*)d_in[1];
  const float* xt   = (const float*)d_in[2];
  float* q    = (float*)d_out;
  float* coef = (float*)d_ws;

  const int d  = in_sizes[0];            // 2048
  const int bs = in_sizes[2] / d;        // 4096
  const int nb = d / BM;                 // 16

  dim3 b1(256), g1((unsigned)((bs + 7) / 8));
  vaup_coef_kernel<<<g1, b1, 0, stream>>>(xt, mean, coef, bs, d);

  dim3 b2(256), g2((unsigned)(nb * (nb + 1) / 2));   // upper-triangular blocks only
  vaup_syrk_kernel<<<g2, b2, 0, stream>>>(xt, va, coef, q, bs, d, nb);
}